// ItemConv_63307817943427
// MI455X (gfx1250) — compile-verified
//
#include <hip/hip_runtime.h>

#define D 100
#define NLAYERS 2

typedef __attribute__((ext_vector_type(2))) float v2f;
typedef __attribute__((ext_vector_type(8))) float v8f;

// avg = features/3 ; deg = 1 (self loop)
__global__ void init_kernel(const float* __restrict__ feat, float* __restrict__ avg,
                            int* __restrict__ deg, int N) {
    int i = blockIdx.x * blockDim.x + threadIdx.x;
    const float inv3 = 1.0f / (float)(NLAYERS + 1);
    if (i < N * D) avg[i] = feat[i] * inv3;
    if (i < N) deg[i] = 1;
}

__global__ void zero_kernel(float* __restrict__ p, int n) {
    int i = blockIdx.x * blockDim.x + threadIdx.x;
    if (i < n) p[i] = 0.0f;
}

__global__ void deg_kernel(const int* __restrict__ dst, int* __restrict__ deg, int E) {
    int e = blockIdx.x * blockDim.x + threadIdx.x;
    if (e < E) atomicAdd(&deg[dst[e]], 1);
}

// y[N,100] = x[N,100] @ Wl[100,100]^T  via V_WMMA_F32_16X16X4_F32
// One wave per 16-row tile; loops 7 column tiles (112 padded cols).
__global__ __launch_bounds__(256)
void gemm_kernel(const float* __restrict__ x, const float* __restrict__ Wl,
                 float* __restrict__ y, int N) {
    __shared__ float sW[112 * 100];   // sW[j*100+k] = W[j][k], rows 100..111 zero
    for (int i = threadIdx.x; i < 112 * 100; i += 256) {
        int j = i / 100, k = i - j * 100;
        sW[i] = (j < 100) ? Wl[j * 100 + k] : 0.0f;
    }
    __syncthreads();

    const int wave = threadIdx.x >> 5;
    const int lane = threadIdx.x & 31;
    const int ntiles = (N + 15) >> 4;
    const int tile = blockIdx.x * 8 + wave;   // wave-uniform predicate: EXEC stays full
    if (tile >= ntiles) return;

    const int row0 = tile * 16;
    const int lm = lane & 15;
    const int kb = (lane < 16) ? 0 : 2;       // 32-bit A 16x4 layout: K = 4*ks + kb, kb+1
    const int arow = row0 + lm;

    // Preload all A fragments for this row tile (reused across 7 col tiles)
    v2f afrag[25];
#pragma unroll
    for (int ks = 0; ks < 25; ++ks) {
        if (arow < N)
            afrag[ks] = *(const v2f*)(x + (size_t)arow * D + ks * 4 + kb);
        else
            afrag[ks] = v2f{0.0f, 0.0f};
    }

    for (int ct = 0; ct < 7; ++ct) {
        const int col0 = ct * 16;
        const int jj = col0 + lm;             // column of output = row of W
        v8f acc = {};
#pragma unroll
        for (int ks = 0; ks < 25; ++ks) {
            // B 4x16 fragment: B[k][n] = W[n][k]
            v2f b = *(const v2f*)(&sW[jj * 100 + ks * 4 + kb]);
            acc = __builtin_amdgcn_wmma_f32_16x16x4_f32(
                false, afrag[ks], false, b, (short)0, acc, false, false);
        }
        // C/D layout: VGPR v -> row0 + v + (lane<16 ? 0 : 8), col = col0 + lane%16
        const int col = col0 + lm;
        if (col < D) {
            const int rbase = row0 + ((lane < 16) ? 0 : 8);
#pragma unroll
            for (int v = 0; v < 8; ++v) {
                int r = rbase + v;
                if (r < N) y[(size_t)r * D + col] = acc[v];
            }
        }
    }
}

// One wave per edge: acc[dst] += y[src] (unweighted; deg_inv applied per-row later)
__global__ __launch_bounds__(256)
void scatter_kernel(const float* __restrict__ y, const int* __restrict__ src,
                    const int* __restrict__ dst, float* __restrict__ acc, int E) {
    int wid = (blockIdx.x * blockDim.x + threadIdx.x) >> 5;
    int lane = threadIdx.x & 31;
    if (wid >= E) return;
    int s = src[wid];
    int d = dst[wid];
    if (lane < 25) {                           // 25 float4 chunks = 100 floats
        const float4 v = *(const float4*)(y + (size_t)s * D + lane * 4);
        float* o = acc + (size_t)d * D + lane * 4;
        atomicAdd(o + 0, v.x);
        atomicAdd(o + 1, v.y);
        atomicAdd(o + 2, v.z);
        atomicAdd(o + 3, v.w);
    }
}

// One wave per node: x = deg_inv*(acc + y_self); avg += normalize(x)/3; xnext = x
__global__ __launch_bounds__(256)
void post_kernel(const float* __restrict__ y, const float* __restrict__ acc,
                 const int* __restrict__ deg, float* __restrict__ xnext,
                 float* __restrict__ avg, int N) {
    int n = (blockIdx.x * blockDim.x + threadIdx.x) >> 5;
    int lane = threadIdx.x & 31;
    if (n >= N) return;
    const float dinv = 1.0f / (float)deg[n];
    float4 v = {0.0f, 0.0f, 0.0f, 0.0f};
    float ss = 0.0f;
    const size_t base = (size_t)n * D + lane * 4;
    if (lane < 25) {
        float4 a = *(const float4*)(acc + base);
        float4 b = *(const float4*)(y + base);   // self-loop contribution
        v.x = (a.x + b.x) * dinv;
        v.y = (a.y + b.y) * dinv;
        v.z = (a.z + b.z) * dinv;
        v.w = (a.w + b.w) * dinv;
        ss = v.x * v.x + v.y * v.y + v.z * v.z + v.w * v.w;
    }
#pragma unroll
    for (int off = 16; off; off >>= 1)          // wave32 allreduce
        ss += __shfl_xor(ss, off, 32);
    const float sc = (1.0f / (float)(NLAYERS + 1)) / fmaxf(sqrtf(ss), 1e-12f);
    if (lane < 25) {
        *(float4*)(xnext + base) = v;
        float4 o = *(const float4*)(avg + base);
        o.x += v.x * sc;
        o.y += v.y * sc;
        o.z += v.z * sc;
        o.w += v.w * sc;
        *(float4*)(avg + base) = o;
    }
}

extern "C" void kernel_launch(void* const* d_in, const int* in_sizes, int n_in,
                              void* d_out, int out_size, void* d_ws, size_t ws_size,
                              hipStream_t stream) {
    const float* feat = (const float*)d_in[0];
    const float* W    = (const float*)d_in[1];
    const int*   src  = (const int*)d_in[2];
    const int*   dst  = (const int*)d_in[3];
    float* avg = (float*)d_out;

    const int N = in_sizes[0] / D;
    const int E = in_sizes[2];
    const int total = N * D;

    char* w = (char*)d_ws;
    auto carve = [&](size_t bytes) -> char* {
        char* p = w;
        w += (bytes + 255) & ~(size_t)255;
        return p;
    };
    int*   deg  = (int*)carve((size_t)N * sizeof(int));
    float* y    = (float*)carve((size_t)total * sizeof(float));
    float* acc  = (float*)carve((size_t)total * sizeof(float));
    float* xcur = (float*)carve((size_t)total * sizeof(float));
    (void)ws_size; (void)n_in; (void)out_size;

    init_kernel<<<(total + 255) / 256, 256, 0, stream>>>(feat, avg, deg, N);
    deg_kernel<<<(E + 255) / 256, 256, 0, stream>>>(dst, deg, E);

    const int ntiles = (N + 15) / 16;
    for (int l = 0; l < NLAYERS; ++l) {
        const float* xin = (l == 0) ? feat : xcur;
        gemm_kernel<<<(ntiles + 7) / 8, 256, 0, stream>>>(xin, W + (size_t)l * D * D, y, N);
        zero_kernel<<<(total + 255) / 256, 256, 0, stream>>>(acc, total);
        scatter_kernel<<<(E + 7) / 8, 256, 0, stream>>>(y, src, dst, acc, E);
        post_kernel<<<(N + 7) / 8, 256, 0, stream>>>(y, acc, deg, xcur, avg, N);
    }
}